// attention_module_25984552141154
// MI455X (gfx1250) — compile-verified
//
#include <hip/hip_runtime.h>

typedef float v2f __attribute__((ext_vector_type(2)));
typedef float v8f __attribute__((ext_vector_type(8)));

#define S_DIM 640
#define B_DIM 8
#define C_DIM 256
#define T_DIM 16
#define K_DIM 16
#define E_DIM 128
#define P_DIM 256

#define NTHREADS 512          // 16 wave32 waves
#define NWAVES   (NTHREADS / 32)

__global__ __launch_bounds__(NTHREADS) void fused_scanpath_attn(
    const float* __restrict__ features,   // (S, B, C)
    const float* __restrict__ attn,       // (B, T, K, S)
    const float* __restrict__ subj,       // (B, E)
    const float* __restrict__ Wf,         // (P, C)
    const float* __restrict__ bf,         // (P)
    const float* __restrict__ Ws,         // (P, E)
    const float* __restrict__ bs,         // (P)
    const float* __restrict__ Wa,         // (1, P)
    const float* __restrict__ ba,         // (1)
    float* __restrict__ out)              // (B, T, K)
{
    __shared__ float A_s[16 * S_DIM];     // 40 KB: attention strip (16 x 640)
    __shared__ float H_s[16 * C_DIM];     // 16 KB: aggr (16 x 256)
    __shared__ float bias_s[P_DIM];       //  1 KB: b_f + b_s + subj @ Ws^T
    __shared__ float logit_s[16];

    const int bt   = blockIdx.x;          // 0..127 : (b,t)
    const int b    = bt >> 4;
    const int tid  = threadIdx.x;         // 0..511 (16 waves of 32)
    const int wave = tid >> 5;            // 0..15
    const int lane = tid & 31;
    const int lo   = lane & 15;
    const int hi   = lane >> 4;           // 0 or 1

    // ---- cooperative load of the 16x640 attention strip into LDS ----
    const float* arow = attn + (size_t)bt * (K_DIM * S_DIM);
    #pragma unroll
    for (int i = 0; i < (16 * S_DIM) / NTHREADS; ++i)
        A_s[i * NTHREADS + tid] = arow[i * NTHREADS + tid];

    // ---- bias[p] = b_f[p] + b_s[p] + subj[b] . Ws[p,:]  (one p per thread) ----
    if (tid < P_DIM) {
        float acc = bf[tid] + bs[tid];
        const float* se = subj + b * E_DIM;
        const float* wr = Ws + (size_t)tid * E_DIM;
        #pragma unroll 4
        for (int e = 0; e < E_DIM; ++e) acc += se[e] * wr[e];
        bias_s[tid] = acc;
    }
    if (tid < 16) logit_s[tid] = ba[0];

    __syncthreads();

    // ================= GEMM1: aggr = Amap(16x640) @ F_b(640x256) / 640 =====
    // each wave owns one 16-wide column tile: cols [n0, n0+16)
    const int n0 = wave * 16;
    v8f acc0 = {};

    const size_t frow = (size_t)B_DIM * C_DIM;          // feature row stride (floats)
    const float* fb0 = features + (size_t)(2 * hi) * frow + b * C_DIM + (n0 + lo);

    #pragma unroll 4
    for (int k = 0; k < S_DIM; k += 4) {
        // A frag: {Amap[lo, k+2hi], Amap[lo, k+2hi+1]}  (8B-aligned LDS read)
        v2f a = *(const v2f*)(&A_s[lo * S_DIM + k + 2 * hi]);
        const float* p = fb0 + (size_t)k * frow;
        v2f b0v;
        b0v.x = p[0];            // F[k+2hi,   n0+lo]
        b0v.y = p[frow];         // F[k+2hi+1, n0+lo]
        acc0 = __builtin_amdgcn_wmma_f32_16x16x4_f32(false, a, false, b0v,
                                                     (short)0, acc0, false, false);
    }

    const float inv_s = 1.0f / (float)S_DIM;
    // D layout: lane holds (m = v + 8*hi, n = n0 + lo) for vgpr v
    #pragma unroll
    for (int v = 0; v < 8; ++v)
        H_s[(v + 8 * hi) * C_DIM + n0 + lo] = acc0[v] * inv_s;
    __syncthreads();

    // ================= GEMM2: proj = aggr(16x256) @ Wf^T ====================
    // each wave owns output cols p in [n0, n0+16)
    v8f p0 = {};
    const float* w0 = Wf + (size_t)(n0 + lo) * C_DIM;        // row p = n0+lo

    #pragma unroll 4
    for (int c = 0; c < C_DIM; c += 4) {
        v2f a   = *(const v2f*)(&H_s[lo * C_DIM + c + 2 * hi]);
        v2f b0v = *(const v2f*)(&w0[c + 2 * hi]);            // contiguous 8B
        p0 = __builtin_amdgcn_wmma_f32_16x16x4_f32(false, a, false, b0v,
                                                   (short)0, p0, false, false);
    }

    // ---- epilogue: h = tanh(proj + bias); logit partials -------------------
    const float bias0 = bias_s[n0 + lo];
    const float wa0   = Wa[n0 + lo];

    #pragma unroll
    for (int v = 0; v < 8; ++v) {
        float h0   = tanhf(p0[v] + bias0);
        float part = h0 * wa0;                   // this wave's 16 p's for m = v+8*hi
        // reduce over the 16 lanes that share (hi, v)
        part += __shfl_xor(part, 1, 32);
        part += __shfl_xor(part, 2, 32);
        part += __shfl_xor(part, 4, 32);
        part += __shfl_xor(part, 8, 32);
        if (lo == 0)                              // lanes 0 (m=v) and 16 (m=v+8)
            atomicAdd(&logit_s[v + 8 * hi], part);
    }
    __syncthreads();

    // ---- softmax over K=16, one wave (lanes 16-31 mirror lanes 0-15) -------
    if (tid < 32) {
        float l = logit_s[lane & 15];
        float m = l;
        m = fmaxf(m, __shfl_xor(m, 1, 32));
        m = fmaxf(m, __shfl_xor(m, 2, 32));
        m = fmaxf(m, __shfl_xor(m, 4, 32));
        m = fmaxf(m, __shfl_xor(m, 8, 32));
        float e = __expf(l - m);
        float s = e;
        s += __shfl_xor(s, 1, 32);
        s += __shfl_xor(s, 2, 32);
        s += __shfl_xor(s, 4, 32);
        s += __shfl_xor(s, 8, 32);
        if (lane < 16)
            out[(size_t)bt * K_DIM + lane] = e / s;
    }
}

extern "C" void kernel_launch(void* const* d_in, const int* in_sizes, int n_in,
                              void* d_out, int out_size, void* d_ws, size_t ws_size,
                              hipStream_t stream) {
    (void)in_sizes; (void)n_in; (void)out_size; (void)d_ws; (void)ws_size;
    const float* features = (const float*)d_in[0];
    const float* attn     = (const float*)d_in[1];
    const float* subj     = (const float*)d_in[2];
    const float* Wf       = (const float*)d_in[3];
    const float* bf       = (const float*)d_in[4];
    const float* Ws       = (const float*)d_in[5];
    const float* bs       = (const float*)d_in[6];
    const float* Wa       = (const float*)d_in[7];
    const float* ba       = (const float*)d_in[8];
    float* out = (float*)d_out;

    dim3 grid(B_DIM * T_DIM);   // 128 workgroups: one per (b, t)
    dim3 block(NTHREADS);       // 16 wave32 waves
    hipLaunchKernelGGL(fused_scanpath_attn, grid, block, 0, stream,
                       features, attn, subj, Wf, bf, Ws, bs, Wa, ba, out);
}